// GCNEncoder_94489281077
// MI455X (gfx1250) — compile-verified
//
#include <hip/hip_runtime.h>

typedef __attribute__((ext_vector_type(2))) float v2f;
typedef __attribute__((ext_vector_type(8))) float v8f;

#define IN_DIM  128
#define HID_DIM 128
#define OUT_DIM 64

static __device__ __forceinline__ void atomic_add_f32(float* p, float v) {
  __hip_atomic_fetch_add(p, v, __ATOMIC_RELAXED, __HIP_MEMORY_SCOPE_AGENT);
}

// ---------------- degree / norm ----------------

__global__ void k_deg_init(float* __restrict__ deg, int n) {
  int i = blockIdx.x * blockDim.x + threadIdx.x;
  if (i < n) deg[i] = 1.0f;  // self-loop contributes 1
}

__global__ void k_deg_accum(const long long* __restrict__ dst, float* __restrict__ deg, int e) {
  int i = blockIdx.x * blockDim.x + threadIdx.x;
  if (i < e) atomic_add_f32(&deg[(int)dst[i]], 1.0f);
}

__global__ void k_rsqrt(float* __restrict__ deg, int n) {
  int i = blockIdx.x * blockDim.x + threadIdx.x;
  if (i < n) deg[i] = rsqrtf(deg[i]);
}

__global__ void k_norm(const long long* __restrict__ src, const long long* __restrict__ dst,
                       const float* __restrict__ dinv, float* __restrict__ norm, int e) {
  int i = blockIdx.x * blockDim.x + threadIdx.x;
  if (i < e) norm[i] = dinv[(int)src[i]] * dinv[(int)dst[i]];
}

// ---------------- WMMA GEMM: D[nrows x NCOL] = A[nrows x K] * B[K x NCOL] ----------------
// One wave32 computes one 16x16 output tile via V_WMMA_F32_16X16X4_F32 (fp32 matrix pipe).
// A layout (16x4):  lanes 0-15 hold M=lane (K=k+0 in a.x, k+1 in a.y),
//                   lanes 16-31 hold M=lane-16 (K=k+2 in a.x, k+3 in a.y).
// B layout (4x16):  vgpr0 = rows K=k+0 (lanes 0-15) / K=k+2 (lanes 16-31), cols N=lane%16.
// C/D layout:       vgpr i: lanes 0-15 -> M=i, lanes 16-31 -> M=8+i; N=lane%16.
template <int K, int NCOL>
__global__ void k_gemm_wmma(const float* __restrict__ A, const float* __restrict__ B,
                            float* __restrict__ D, int nrows) {
  const int wave = (blockIdx.x * blockDim.x + threadIdx.x) >> 5;
  const int lane = threadIdx.x & 31;
  const int col_tiles = NCOL / 16;
  const int row_base = (wave / col_tiles) * 16;
  const int col_base = (wave % col_tiles) * 16;
  if (row_base >= nrows) return;  // wave-uniform: EXEC stays all-1s for WMMA

  const int m  = lane & 15;   // row in A-tile / col in B,C,D-tile
  const int kh = lane >> 4;   // 0 -> K offsets {0,1}; 1 -> {2,3}

  const float* arow = A + (size_t)(row_base + m) * K;
  v8f c = {};
#pragma unroll 8
  for (int k = 0; k < K; k += 4) {
    v2f a, b;
    a.x = arow[k + 2 * kh + 0];
    a.y = arow[k + 2 * kh + 1];
    b.x = B[(size_t)(k + 2 * kh + 0) * NCOL + col_base + m];
    b.y = B[(size_t)(k + 2 * kh + 1) * NCOL + col_base + m];
    c = __builtin_amdgcn_wmma_f32_16x16x4_f32(false, a, false, b, (short)0, c, false, false);
  }
#pragma unroll
  for (int i = 0; i < 8; ++i) {
    D[(size_t)(row_base + i + 8 * kh) * NCOL + col_base + m] = c[i];
  }
}

// ---------------- self-loop init: agg[i] = dinv[i]^2 * h[i] (+ bias) ----------------

template <int FEAT>
__global__ void k_self_init(const float* __restrict__ h, const float* __restrict__ dinv,
                            const float* __restrict__ bias, float* __restrict__ agg, int n) {
  const int F4 = FEAT / 4;
  int gid = blockIdx.x * blockDim.x + threadIdx.x;
  int i = gid / F4, q = gid % F4;
  if (i >= n) return;
  float di = dinv[i];
  float w = di * di;
  const float4 hv = *(const float4*)(h + (size_t)i * FEAT + q * 4);
  float4 r;
  r.x = w * hv.x; r.y = w * hv.y; r.z = w * hv.z; r.w = w * hv.w;
  if (bias) {
    const float4 b = *(const float4*)(bias + q * 4);
    r.x += b.x; r.y += b.y; r.z += b.z; r.w += b.w;
  }
  *(float4*)(agg + (size_t)i * FEAT + q * 4) = r;
}

// ---------------- per-edge scatter: agg[dst] += norm * h[src] (L2-resident atomics) ----------------

template <int FEAT>
__global__ void k_scatter(const float* __restrict__ h, const long long* __restrict__ esrc,
                          const long long* __restrict__ edst, const float* __restrict__ norm,
                          float* __restrict__ agg, int nedges) {
  const int F4 = FEAT / 4;
  long long gid = (long long)blockIdx.x * blockDim.x + threadIdx.x;
  int e = (int)(gid / F4);
  int q = (int)(gid % F4);
  if (e >= nedges) return;
  int s = (int)esrc[e];
  int d = (int)edst[e];
  float w = norm[e];
  const float4 hv = *(const float4*)(h + (size_t)s * FEAT + q * 4);
  float* out = agg + (size_t)d * FEAT + q * 4;
  atomic_add_f32(out + 0, w * hv.x);
  atomic_add_f32(out + 1, w * hv.y);
  atomic_add_f32(out + 2, w * hv.z);
  atomic_add_f32(out + 3, w * hv.w);
}

// ---------------- bias + ReLU (layer-1 epilogue) ----------------

__global__ void k_bias_relu(const float* __restrict__ agg, const float* __restrict__ bias,
                            float* __restrict__ out, int n) {
  const int F4 = HID_DIM / 4;
  int gid = blockIdx.x * blockDim.x + threadIdx.x;
  int i = gid / F4, q = gid % F4;
  if (i >= n) return;
  const float4 a = *(const float4*)(agg + (size_t)i * HID_DIM + q * 4);
  const float4 b = *(const float4*)(bias + q * 4);
  float4 r;
  r.x = fmaxf(a.x + b.x, 0.0f);
  r.y = fmaxf(a.y + b.y, 0.0f);
  r.z = fmaxf(a.z + b.z, 0.0f);
  r.w = fmaxf(a.w + b.w, 0.0f);
  *(float4*)(out + (size_t)i * HID_DIM + q * 4) = r;
}

// ---------------- launch ----------------

extern "C" void kernel_launch(void* const* d_in, const int* in_sizes, int n_in,
                              void* d_out, int out_size, void* d_ws, size_t ws_size,
                              hipStream_t stream) {
  const float*     x   = (const float*)d_in[0];
  const long long* ei  = (const long long*)d_in[1];  // int64 edge_index [2, E]
  const float*     W1  = (const float*)d_in[2];
  const float*     b1  = (const float*)d_in[3];
  const float*     W2  = (const float*)d_in[4];
  const float*     b2  = (const float*)d_in[5];
  float*           out = (float*)d_out;

  const int N = in_sizes[0] / IN_DIM;
  const int E = in_sizes[1] / 2;
  const long long* src = ei;
  const long long* dst = ei + E;

  // workspace carve-up (256B aligned)
  char*  ws  = (char*)d_ws;
  size_t off = 0;
  auto alloc = [&](size_t bytes) -> float* {
    float* p = (float*)(ws + off);
    off += (bytes + 255) & ~(size_t)255;
    return p;
  };
  float* dinv = alloc((size_t)N * 4);            // degree -> rsqrt(degree)
  float* norm = alloc((size_t)E * 4);            // per-edge norm
  float* h    = alloc((size_t)N * HID_DIM * 4);  // x@W1, later relu(agg1+b1)
  float* agg  = alloc((size_t)N * HID_DIM * 4);  // layer-1 aggregation
  float* h2   = alloc((size_t)N * OUT_DIM * 4);  // relu(...)@W2
  (void)ws_size; (void)n_in; (void)out_size;

  const int BS = 256;
  auto blocks = [](long long work, int bs) { return (unsigned)((work + bs - 1) / bs); };

  // 1) normalization
  k_deg_init<<<blocks(N, BS), BS, 0, stream>>>(dinv, N);
  k_deg_accum<<<blocks(E, BS), BS, 0, stream>>>(dst, dinv, E);
  k_rsqrt<<<blocks(N, BS), BS, 0, stream>>>(dinv, N);
  k_norm<<<blocks(E, BS), BS, 0, stream>>>(src, dst, dinv, norm, E);

  // 2) layer 1: h = x @ W1  (WMMA fp32), then aggregate
  {
    long long tiles = (long long)(N / 16) * (HID_DIM / 16);
    k_gemm_wmma<IN_DIM, HID_DIM><<<blocks(tiles * 32, BS), BS, 0, stream>>>(x, W1, h, N);
  }
  k_self_init<HID_DIM><<<blocks((long long)N * (HID_DIM / 4), BS), BS, 0, stream>>>(
      h, dinv, nullptr, agg, N);
  k_scatter<HID_DIM><<<blocks((long long)E * (HID_DIM / 4), BS), BS, 0, stream>>>(
      h, src, dst, norm, agg, E);
  k_bias_relu<<<blocks((long long)N * (HID_DIM / 4), BS), BS, 0, stream>>>(agg, b1, h, N);

  // 3) layer 2: h2 = relu(...) @ W2  (WMMA fp32), then aggregate into d_out
  {
    long long tiles = (long long)(N / 16) * (OUT_DIM / 16);
    k_gemm_wmma<HID_DIM, OUT_DIM><<<blocks(tiles * 32, BS), BS, 0, stream>>>(h, W2, h2, N);
  }
  k_self_init<OUT_DIM><<<blocks((long long)N * (OUT_DIM / 4), BS), BS, 0, stream>>>(
      h2, dinv, b2, out, N);
  k_scatter<OUT_DIM><<<blocks((long long)E * (OUT_DIM / 4), BS), BS, 0, stream>>>(
      h2, src, dst, norm, out, E);
}